// TaggingFNNCRFDecoder_10849087390247
// MI455X (gfx1250) — compile-verified
//
#include <hip/hip_runtime.h>
#include <hip/hip_bf16.h>
#include <stdint.h>

// Problem constants (from reference): B=128, T=512, H=1024, K=128
#define BB   128
#define TT   512
#define HH   1024
#define KK   128

typedef float v2f __attribute__((ext_vector_type(2)));
typedef float v8f __attribute__((ext_vector_type(8)));

// ---------------------------------------------------------------------------
// Kernel 1: logits[b,t,k] = sum_h hiddens[b,t,h] * W[k,h] + bias[k]
// M = B*T = 65536 rows. Each wave computes a 16x128 strip via
// V_WMMA_F32_16X16X4_F32, accumulating over H/4 = 256 steps, 8 N-tiles.
// A fragment (16x4 f32): lane<16 -> row=lane, K=0,1 ; lane>=16 -> row=lane-16,
// K=2,3  => per-lane float2 at hiddens[row*H + h + koff].
// B fragment (4x16 f32): v0 lanes0-15 = K-row 0, lanes16-31 = K-row 2 (mirrors
// C/D striping) => per-lane float2 at W[n*H + h + koff].
// C/D (16x16 f32): VGPR r -> M = r (lanes 0-15) / r+8 (lanes 16-31), N = lane%16.
// ---------------------------------------------------------------------------
__global__ __launch_bounds__(128) void crf_logits_wmma(
    const float* __restrict__ hiddens,   // [M, H]
    const float* __restrict__ W,         // [K, H]
    const float* __restrict__ bias,      // [K]
    float* __restrict__ logits)          // [M, K]
{
  const int lane  = threadIdx.x & 31;
  const int wave  = threadIdx.x >> 5;
  const long strip = (long)blockIdx.x * 4 + wave;   // 4096 strips total
  const long m0    = strip * 16;

  const int row  = lane & 15;
  const int koff = (lane >> 4) << 1;                // 0 or 2

  const float* aptr = hiddens + (m0 + row) * HH + koff;
  const float* bptr = W + (long)(lane & 15) * HH + koff;

  v8f acc[8];
  #pragma unroll
  for (int nt = 0; nt < 8; ++nt) { v8f z = {}; acc[nt] = z; }

  #pragma unroll 2
  for (int h = 0; h < HH; h += 4) {
    v2f a = *(const v2f*)(aptr + h);
    #pragma unroll
    for (int nt = 0; nt < 8; ++nt) {
      v2f bf = *(const v2f*)(bptr + (long)nt * 16 * HH + h);
      acc[nt] = __builtin_amdgcn_wmma_f32_16x16x4_f32(
          /*neg_a=*/false, a, /*neg_b=*/false, bf,
          /*c_mod=*/(short)0, acc[nt], /*reuse_a=*/false, /*reuse_b=*/false);
    }
  }

  const int mrow = (lane >> 4) << 3;                // 0 or 8
  #pragma unroll
  for (int nt = 0; nt < 8; ++nt) {
    const int n = nt * 16 + (lane & 15);
    const float bv = bias[n];
    #pragma unroll
    for (int r = 0; r < 8; ++r) {
      const long m = m0 + mrow + r;
      logits[m * KK + n] = acc[nt][r] + bv;
    }
  }
}

// ---------------------------------------------------------------------------
// Kernel 2: forward log-partition + Viterbi + backtrack, one block per batch.
// 256 threads: j = tid & 127 (next tag), seg = tid >> 7 (prev-tag half).
// LDS: trans (64KB) + backpointer history (64KB u8) + alpha/score + reduction.
// ---------------------------------------------------------------------------
#define SM_TRANS 0
#define SM_HIST  (65536)
#define SM_ALPHA (SM_HIST + TT * KK)        // 131072
#define SM_SCORE (SM_ALPHA + KK * 4)
#define SM_RV    (SM_SCORE + KK * 4)        // 256 floats: seg vmax
#define SM_RS    (SM_RV + 256 * 4)          // 256 floats: seg exp-sum
#define SM_RW    (SM_RS + 256 * 4)          // 256 floats: seg viterbi max
#define SM_RA    (SM_RW + 256 * 4)          // 256 ints  : seg viterbi argmax
#define SM_TAGS  (SM_RA + 256 * 4)          // 512 ints
#define SM_TOTAL (SM_TAGS + TT * 4)         // 138240 bytes

__global__ __launch_bounds__(256) void crf_forward_viterbi(
    const float* __restrict__ logits,      // [B*T, K]
    const uint8_t* __restrict__ mask,      // [B, T] (bool)
    const float* __restrict__ start_t,     // [K]
    const float* __restrict__ end_t,       // [K]
    const float* __restrict__ trans,       // [K, K]
    float* __restrict__ den_out,           // [B]
    float* __restrict__ pred_out)          // [B, T] as float
{
  extern __shared__ char smem[];
  float*   s_trans = (float*)(smem + SM_TRANS);
  uint8_t* s_hist  = (uint8_t*)(smem + SM_HIST);
  float*   s_alpha = (float*)(smem + SM_ALPHA);
  float*   s_score = (float*)(smem + SM_SCORE);
  float*   s_rv    = (float*)(smem + SM_RV);
  float*   s_rs    = (float*)(smem + SM_RS);
  float*   s_rw    = (float*)(smem + SM_RW);
  int*     s_ra    = (int*)(smem + SM_RA);
  int*     s_tags  = (int*)(smem + SM_TAGS);

  const int b   = blockIdx.x;
  const int tid = threadIdx.x;
  const int j   = tid & (KK - 1);
  const int seg = tid >> 7;                 // 0 or 1
  const int ibase = seg * 64;

  // stage transition matrix into LDS
  for (int idx = tid; idx < KK * KK; idx += 256) s_trans[idx] = trans[idx];

  if (seg == 0) {
    const float e0 = logits[((long)b * TT) * KK + j];
    s_alpha[j] = start_t[j] + e0;
    s_score[j] = start_t[j] + e0;
  }
  __syncthreads();

  for (int t = 1; t < TT; ++t) {
    const float e  = logits[((long)b * TT + t) * KK + j];
    const int   mt = mask[(long)b * TT + t];

    // pass 1: max over this segment's 64 prev tags (lse max + viterbi max/arg)
    float vmax = -3.4e38f, wmax = -3.4e38f;
    int warg = ibase;
    #pragma unroll 4
    for (int ii = 0; ii < 64; ++ii) {
      const int i = ibase + ii;
      const float tr = s_trans[i * KK + j];
      const float v  = s_alpha[i] + tr;
      const float w  = s_score[i] + tr;
      vmax = fmaxf(vmax, v);
      if (w > wmax) { wmax = w; warg = i; }
    }
    // pass 2: stabilized exp-sum for logsumexp
    float ssum = 0.0f;
    #pragma unroll 4
    for (int ii = 0; ii < 64; ++ii) {
      const int i = ibase + ii;
      const float v = s_alpha[i] + s_trans[i * KK + j];
      ssum += __expf(v - vmax);
    }

    s_rv[tid] = vmax; s_rs[tid] = ssum; s_rw[tid] = wmax; s_ra[tid] = warg;
    __syncthreads();

    float anew = 0.0f, snew = 0.0f; int hnew = 0;
    if (seg == 0) {
      const float vm1 = s_rv[j + 128], s1 = s_rs[j + 128];
      const float wm1 = s_rw[j + 128]; const int wa1 = s_ra[j + 128];
      const float vm = fmaxf(vmax, vm1);
      const float st = ssum * __expf(vmax - vm) + s1 * __expf(vm1 - vm);
      anew = vm + __logf(st) + e;
      if (wm1 > wmax) { wmax = wm1; warg = wa1; }   // strict >: first index wins
      snew = wmax + e;
      hnew = warg;
    }
    __syncthreads();
    if (seg == 0) {
      if (mt) { s_alpha[j] = anew; s_score[j] = snew; }
      s_hist[t * KK + j] = (uint8_t)hnew;
    }
    __syncthreads();
  }

  // final terms
  if (seg == 0) {
    s_rv[j] = s_alpha[j] + end_t[j];      // for log-partition
    s_rw[j] = s_score[j] + end_t[j];      // for viterbi last tag
  }
  __syncthreads();

  if (tid == 0) {
    float vm = -3.4e38f, wm = -3.4e38f; int wa = 0;
    for (int i = 0; i < KK; ++i) {
      vm = fmaxf(vm, s_rv[i]);
      if (s_rw[i] > wm) { wm = s_rw[i]; wa = i; }
    }
    float st = 0.0f;
    for (int i = 0; i < KK; ++i) st += __expf(s_rv[i] - vm);
    den_out[b] = vm + __logf(st);

    // backtrack entirely in LDS
    int cur = wa;
    s_tags[TT - 1] = cur;
    for (int t = TT - 1; t >= 1; --t) {
      if (mask[(long)b * TT + t]) cur = (int)s_hist[t * KK + cur];
      s_tags[t - 1] = cur;
    }
  }
  __syncthreads();

  for (int t = tid; t < TT; t += 256)
    pred_out[(long)b * TT + t] = (float)s_tags[t];
}

// ---------------------------------------------------------------------------
// Kernel 3: numerator score per batch element.
// ---------------------------------------------------------------------------
__global__ __launch_bounds__(128) void crf_numerator(
    const float* __restrict__ logits,
    const uint8_t* __restrict__ mask,
    const int* __restrict__ labels,
    const float* __restrict__ start_t,
    const float* __restrict__ end_t,
    const float* __restrict__ trans,
    float* __restrict__ num_out)
{
  const int b = blockIdx.x, tid = threadIdx.x;
  __shared__ float red[128];
  __shared__ int   cnt[128];

  float acc = 0.0f; int c = 0;
  for (int t = tid; t < TT; t += 128) {
    const int mt = mask[(long)b * TT + t];
    c += mt;
    if (t >= 1 && mt) {
      const int lp = labels[(long)b * TT + t - 1];
      const int lc = labels[(long)b * TT + t];
      acc += trans[lp * KK + lc] + logits[((long)b * TT + t) * KK + lc];
    }
  }
  red[tid] = acc; cnt[tid] = c;
  __syncthreads();
  if (tid == 0) {
    float s = 0.0f; int len = 0;
    for (int i = 0; i < 128; ++i) { s += red[i]; len += cnt[i]; }
    const int l0 = labels[(long)b * TT];
    s += start_t[l0] + logits[((long)b * TT) * KK + l0];
    const int ll = labels[(long)b * TT + (len - 1)];
    s += end_t[ll];
    num_out[b] = s;
  }
}

// ---------------------------------------------------------------------------
// Kernel 4: loss = -mean(num - den)
// ---------------------------------------------------------------------------
__global__ __launch_bounds__(128) void crf_loss(
    const float* __restrict__ num, const float* __restrict__ den,
    float* __restrict__ out_loss)
{
  __shared__ float red[128];
  const int tid = threadIdx.x;
  red[tid] = num[tid] - den[tid];
  __syncthreads();
  if (tid == 0) {
    float s = 0.0f;
    for (int i = 0; i < BB; ++i) s += red[i];
    *out_loss = -s / (float)BB;
  }
}

// ---------------------------------------------------------------------------
extern "C" void kernel_launch(void* const* d_in, const int* in_sizes, int n_in,
                              void* d_out, int out_size, void* d_ws, size_t ws_size,
                              hipStream_t stream) {
  const float*   hiddens = (const float*)d_in[0];
  const uint8_t* mask    = (const uint8_t*)d_in[1];   // jnp bool -> 1 byte
  const int*     labels  = (const int*)d_in[2];
  const float*   W       = (const float*)d_in[3];
  const float*   bias    = (const float*)d_in[4];
  const float*   start_t = (const float*)d_in[5];
  const float*   end_t   = (const float*)d_in[6];
  const float*   trans   = (const float*)d_in[7];

  float* out    = (float*)d_out;                      // pred[B*T] then loss
  float* logits = (float*)d_ws;                       // 32 MB
  float* den    = logits + (size_t)BB * TT * KK;
  float* num    = den + BB;

  // allow >64KB dynamic LDS for the scan kernel (attribute set is not a
  // stream op; safe & idempotent under graph capture)
  (void)hipFuncSetAttribute((const void*)crf_forward_viterbi,
                            hipFuncAttributeMaxDynamicSharedMemorySize,
                            SM_TOTAL);

  // 1) logits GEMM via f32 WMMA: 4096 strips of 16 rows, 4 waves/block
  crf_logits_wmma<<<1024, 128, 0, stream>>>(hiddens, W, bias, logits);

  // 2) forward + viterbi + backtrack (one block per batch element)
  crf_forward_viterbi<<<BB, 256, SM_TOTAL, stream>>>(
      logits, mask, start_t, end_t, trans, den, out);

  // 3) numerator
  crf_numerator<<<BB, 128, 0, stream>>>(logits, mask, labels, start_t, end_t,
                                        trans, num);

  // 4) loss scalar
  crf_loss<<<1, 128, 0, stream>>>(num, den, out + (size_t)BB * TT);
}